// OuterProduct_46505905881344
// MI455X (gfx1250) — compile-verified
//
#include <hip/hip_runtime.h>

typedef __attribute__((ext_vector_type(2))) float v2f;
typedef __attribute__((ext_vector_type(8))) float v8f;

#define NTOK 768
#define DIM  128
#define LSTR 130   // padded LDS row stride (floats): stride%64==2 banks -> conflict-free b64 reads

// ---------------- Kernel 1: left = x @ W_in^T + b_in  (tiny, VALU) ----------------
__global__ __launch_bounds__(128) void proj_in_kernel(
    const float* __restrict__ x, const float* __restrict__ W_in,
    const float* __restrict__ b_in, float* __restrict__ left)
{
  __shared__ float sx[DIM];
  const int i = blockIdx.x;
  const int e = threadIdx.x;
  sx[e] = x[i * DIM + e];
  __syncthreads();
  const float* w = W_in + e * DIM;
  float acc = 0.f;
#pragma unroll 8
  for (int d = 0; d < DIM; ++d) acc = fmaf(sx[d], w[d], acc);
  left[i * DIM + e] = acc + b_in[e];
}

// ---- Kernel 2: out[i,j,e] = sum_d left[i,d]*left[j,d]*W_out[e,d] + b_out[e] ----
// One block per 16x16 (i,j) tile. 8 waves; wave w owns i-rows {2w,2w+1},
// WMMA M-rows are the 16 j's. K=128 swept in 32 steps of V_WMMA_F32_16X16X4_F32.
__global__ __launch_bounds__(256, 1) void pair_gemm_kernel(
    const float* __restrict__ left, const float* __restrict__ W_out,
    const float* __restrict__ b_out, float* __restrict__ out)
{
  extern __shared__ float smem[];
  float* sW  = smem;                  // [128][LSTR]  sW[e][d] = W_out[e][d]
  float* sLi = sW + DIM * LSTR;       // [16][LSTR]   i-rows of left
  float* sLj = sLi + 16 * LSTR;       // [16][LSTR]   j-rows of left

  const int tid = threadIdx.x;
  const int i0 = blockIdx.y * 16;
  const int j0 = blockIdx.x * 16;

  for (int idx = tid; idx < DIM * DIM; idx += 256) {
    const int e = idx >> 7, d = idx & 127;
    sW[e * LSTR + d] = W_out[idx];
  }
  for (int idx = tid; idx < 16 * DIM; idx += 256) {
    const int r = idx >> 7, d = idx & 127;
    sLi[r * LSTR + d] = left[(i0 + r) * DIM + d];
    sLj[r * LSTR + d] = left[(j0 + r) * DIM + d];
  }
  __syncthreads();

  const int wave = tid >> 5;     // wave32
  const int lane = tid & 31;
  const int half = lane >> 4;    // selects K-pair {0,1} vs {2,3} of each 4-wide K step
  const int l15  = lane & 15;    // A: M row (=j offset b); B: N col within tile

  const float* pLi0 = sLi + (wave * 2 + 0) * LSTR;
  const float* pLi1 = sLi + (wave * 2 + 1) * LSTR;
  const float* pLj  = sLj + l15 * LSTR;
  const float* pB   = sW  + l15 * LSTR;   // row n = nt*16 + l15 via +nt*16*LSTR

  v8f acc0[8] = {};   // i = i0 + 2*wave
  v8f acc1[8] = {};   // i = i0 + 2*wave + 1

  for (int k = 0; k < 32; ++k) {
    const int d0 = 4 * k + 2 * half;           // this lane's K pair (d0, d0+1)
    const v2f li0 = *(const v2f*)(pLi0 + d0);  // LDS broadcast within half-wave
    const v2f li1 = *(const v2f*)(pLi1 + d0);
    const v2f lj  = *(const v2f*)(pLj  + d0);
    const v2f a0 = li0 * lj;                   // A[m=b, d] = left_i[d] * left_j[b, d]
    const v2f a1 = li1 * lj;
#pragma unroll
    for (int nt = 0; nt < 8; ++nt) {
      const v2f b = *(const v2f*)(pB + nt * (16 * LSTR) + d0);  // B[d, n] = W_out[n][d]
      acc0[nt] = __builtin_amdgcn_wmma_f32_16x16x4_f32(
          false, a0, false, b, (short)0, acc0[nt], false, false);
      acc1[nt] = __builtin_amdgcn_wmma_f32_16x16x4_f32(
          false, a1, false, b, (short)0, acc1[nt], false, false);
    }
  }

  float bias[8];
#pragma unroll
  for (int nt = 0; nt < 8; ++nt) bias[nt] = b_out[nt * 16 + l15];

  // C/D layout: VGPR v = rows M=v (lanes 0-15) and M=v+8 (lanes 16-31), N = l15.
#pragma unroll
  for (int a = 0; a < 2; ++a) {
    const int i = i0 + wave * 2 + a;
    const v8f* acc = a ? acc1 : acc0;
#pragma unroll
    for (int v = 0; v < 8; ++v) {
      const int j = j0 + v + 8 * half;
      float* po = out + ((size_t)i * NTOK + j) * DIM + l15;
#pragma unroll
      for (int nt = 0; nt < 8; ++nt) {
        po[nt * 16] = acc[nt][v] + bias[nt];
      }
    }
  }
}

extern "C" void kernel_launch(void* const* d_in, const int* in_sizes, int n_in,
                              void* d_out, int out_size, void* d_ws, size_t ws_size,
                              hipStream_t stream)
{
  const float* x     = (const float*)d_in[0];
  const float* W_in  = (const float*)d_in[1];
  const float* b_in  = (const float*)d_in[2];
  const float* W_out = (const float*)d_in[3];
  const float* b_out = (const float*)d_in[4];
  float* out  = (float*)d_out;
  float* left = (float*)d_ws;   // 768*128 floats = 384 KB scratch

  proj_in_kernel<<<NTOK, DIM, 0, stream>>>(x, W_in, b_in, left);

  dim3 grid(NTOK / 16, NTOK / 16);
  const size_t smem = (size_t)(DIM * LSTR + 2 * 16 * LSTR) * sizeof(float);  // 83,200 B
  pair_gemm_kernel<<<grid, 256, smem, stream>>>(left, W_out, b_out, out);
}